// MoELayer_12644383719775
// MI455X (gfx1250) — compile-verified
//
#include <hip/hip_runtime.h>
#include <hip/hip_bf16.h>

// ---------------------------------------------------------------------------
// MoELayer on MI455X (gfx1250, wave32):
//   1) router_partial : scores partial GEMM, K-split x16, fp32 WMMA 16x16x4,
//                       LDS double-buffered so global fetch hides under WMMA
//   2) router_reduce  : deterministic K-slice reduction + bias
//   3) topk_kernel    : per-row top-128-of-512 by |score|, sorted indices
//   4) conv_kernel    : implicit-GEMM 3x3 conv ONLY on the selected channels
//                       (4x less conv work), tap-major K order so address math
//                       is hoisted out of the WMMA loop; 2 channel tiles per
//                       wave share each im2col B fragment (halves x traffic)
// ---------------------------------------------------------------------------

typedef __attribute__((ext_vector_type(2))) float v2f;
typedef __attribute__((ext_vector_type(4))) float v4f;
typedef __attribute__((ext_vector_type(8))) float v8f;

constexpr int BATCH  = 64;
constexpr int CIN    = 64;
constexpr int HW     = 56;
constexpr int NPIX   = HW * HW;          // 3136
constexpr int COUT   = 128;              // kept channels per sample
constexpr int CH     = 512;              // router width (4 experts * 128)
constexpr int IN_DIM = CIN * NPIX;       // 200704
constexpr int KCONV  = CIN * 9;          // 576 (im2col reduction)
constexpr int KSPLIT = 16;               // K-slices for router GEMM
constexpr int KC     = 32;               // K chunk staged through LDS

// ---------------------------------------------------------------------------
// Kernel 1: router partial GEMM, double-buffered through LDS.
// grid = (CH/64, KSPLIT), block = 256 (8 waves).
// Block computes the full 64-row x 64-col score tile for one K slice.
// Each wave owns 2 adjacent 16x16 tiles (shares the A fragment).
// ---------------------------------------------------------------------------
__global__ __launch_bounds__(256) void router_partial(
    const float* __restrict__ x,        // [64, 200704]
    const float* __restrict__ rw,       // [512, 200704]
    float* __restrict__ partial)        // [KSPLIT, 64, 512]
{
    __shared__ float As[2][64][36];     // padded rows (36 words): conflict-free frags
    __shared__ float Bs[2][64][36];

    const int c0      = blockIdx.x * 64;
    const int kslice  = blockIdx.y;
    const int kbeg    = kslice * (IN_DIM / KSPLIT);
    const int nchunks = (IN_DIM / KSPLIT) / KC;   // 392

    const int t    = threadIdx.x;
    const int wave = t >> 5;
    const int lane = t & 31;

    // WMMA fragment decode (16x16x4 f32): A/B are v2f per lane
    const int fm = lane & 15;           // A row / B col
    const int fk = (lane >> 4) * 2;     // K pair select

    // cooperative load mapping: thread -> (row, 4-float column)
    const int lr = t >> 3;              // 0..31
    const int lc = (t & 7) * 4;         // 0,4,...,28

    // wave -> one row tile, two adjacent column tiles (A fragment shared)
    const int tid0 = wave * 2;
    const int rt   = tid0 >> 2;         // 0..3
    const int ct0  = (tid0 & 3);        // 0 or 2
    const int ct1  = ct0 + 1;

    v8f acc0 = {};
    v8f acc1 = {};

    const float* xA0 = x  + (size_t)lr        * IN_DIM + kbeg + lc;
    const float* xA1 = x  + (size_t)(lr + 32) * IN_DIM + kbeg + lc;
    const float* wB0 = rw + (size_t)(c0 + lr)      * IN_DIM + kbeg + lc;
    const float* wB1 = rw + (size_t)(c0 + lr + 32) * IN_DIM + kbeg + lc;

    // prologue: fetch + stage chunk 0
    {
        v4f a0 = *(const v4f*)xA0;
        v4f a1 = *(const v4f*)xA1;
        v4f b0 = __builtin_nontemporal_load((const v4f*)wB0);
        v4f b1 = __builtin_nontemporal_load((const v4f*)wB1);
        *(v4f*)&As[0][lr][lc]      = a0;
        *(v4f*)&As[0][lr + 32][lc] = a1;
        *(v4f*)&Bs[0][lr][lc]      = b0;
        *(v4f*)&Bs[0][lr + 32][lc] = b1;
    }
    __syncthreads();

    int buf = 0;
    for (int c = 0; c < nchunks; ++c) {
        const bool more = (c + 1 < nchunks);
        v4f na0, na1, nb0, nb1;
        if (more) {                      // issue next chunk's loads BEFORE compute
            const int koff = (c + 1) * KC;
            na0 = *(const v4f*)(xA0 + koff);
            na1 = *(const v4f*)(xA1 + koff);
            nb0 = __builtin_nontemporal_load((const v4f*)(wB0 + koff));
            nb1 = __builtin_nontemporal_load((const v4f*)(wB1 + koff));
        }

        #pragma unroll
        for (int kk = 0; kk < KC; kk += 4) {
            v2f a, bA, bB;
            a.x  = As[buf][rt * 16 + fm][kk + fk];
            a.y  = As[buf][rt * 16 + fm][kk + fk + 1];
            bA.x = Bs[buf][ct0 * 16 + fm][kk + fk];
            bA.y = Bs[buf][ct0 * 16 + fm][kk + fk + 1];
            bB.x = Bs[buf][ct1 * 16 + fm][kk + fk];
            bB.y = Bs[buf][ct1 * 16 + fm][kk + fk + 1];
            acc0 = __builtin_amdgcn_wmma_f32_16x16x4_f32(false, a, false, bA,
                                                         (short)0, acc0, false, false);
            acc1 = __builtin_amdgcn_wmma_f32_16x16x4_f32(false, a, false, bB,
                                                         (short)0, acc1, false, false);
        }

        if (more) {                      // stage next chunk into the other buffer
            const int nb = buf ^ 1;
            *(v4f*)&As[nb][lr][lc]      = na0;
            *(v4f*)&As[nb][lr + 32][lc] = na1;
            *(v4f*)&Bs[nb][lr][lc]      = nb0;
            *(v4f*)&Bs[nb][lr + 32][lc] = nb1;
            __syncthreads();             // one barrier per chunk
            buf = nb;
        }
    }

    // store partial tile: D layout M = v + 8*(lane>=16), N = lane&15
    const int mh = (lane >> 4) * 8;
    const int n  = lane & 15;
    float* pbase = partial + (size_t)kslice * BATCH * CH;
    #pragma unroll
    for (int v = 0; v < 8; ++v) {
        const int row = rt * 16 + v + mh;
        pbase[(size_t)row * CH + c0 + ct0 * 16 + n] = acc0[v];
        pbase[(size_t)row * CH + c0 + ct1 * 16 + n] = acc1[v];
    }
}

// ---------------------------------------------------------------------------
// Kernel 2: reduce K slices + bias (deterministic fixed-order sum).
// ---------------------------------------------------------------------------
__global__ __launch_bounds__(256) void router_reduce(
    const float* __restrict__ partial,  // [KSPLIT, 64, 512]
    const float* __restrict__ rb,       // [512]
    float* __restrict__ scores)         // [64, 512]
{
    const int i = blockIdx.x * 256 + threadIdx.x;
    if (i < BATCH * CH) {
        float s = rb[i & (CH - 1)];
        #pragma unroll
        for (int ks = 0; ks < KSPLIT; ++ks)
            s += partial[(size_t)ks * BATCH * CH + i];
        scores[i] = s;
    }
}

// ---------------------------------------------------------------------------
// Kernel 3: top-128-of-512 per row by |score|; output channel-ordered indices.
// Rank = #(strictly greater) + #(equal with lower index)  -> matches
// jax.lax.top_k tie-breaking; keeping rank<128 then compacting in channel
// order == sort(top_k indices).
// ---------------------------------------------------------------------------
__global__ __launch_bounds__(512) void topk_kernel(
    const float* __restrict__ scores,   // [64, 512]
    int* __restrict__ idx)              // [64, 128]
{
    __shared__ float sabs[CH];
    __shared__ int   keep[CH];
    const int b = blockIdx.x;
    const int t = threadIdx.x;

    const float v = fabsf(scores[(size_t)b * CH + t]);
    sabs[t] = v;
    __syncthreads();

    int rank = 0;
    for (int u = 0; u < CH; ++u) {
        const float su = sabs[u];
        rank += (su > v) || (su == v && u < t);
    }
    keep[t] = (rank < COUT) ? 1 : 0;
    __syncthreads();

    if (keep[t]) {
        int pos = 0;
        for (int u = 0; u < t; ++u) pos += keep[u];
        idx[b * COUT + pos] = t;
    }
}

// ---------------------------------------------------------------------------
// Kernel 4: 3x3 conv (pad 1) on selected channels only, implicit GEMM.
// grid = (196 pixel tiles, 4 slot-tile pairs, 64 batches), block = 1 wave.
// Per block: 2 x (16 selected channels) x 16 pixels, K = 576, fp32 WMMA.
// K is enumerated tap-major (k' = tap*64 + ci): the im2col shift and the
// padding predicate are loop-invariant per tap, and both channel tiles share
// every B fragment.
// ---------------------------------------------------------------------------
__global__ __launch_bounds__(32) void conv_kernel(
    const float* __restrict__ x,        // [64, 64, 56, 56]
    const float* __restrict__ w,        // [512, 64, 3, 3] (experts fused)
    const float* __restrict__ bias,     // [512]
    const int*   __restrict__ idx,      // [64, 128]
    float* __restrict__ out)            // [64, 128, 56, 56]
{
    const int pt   = blockIdx.x;        // pixel tile (16 pixels)
    const int mt0  = blockIdx.y * 2;    // first channel-slot tile
    const int mt1  = mt0 + 1;
    const int b    = blockIdx.z;
    const int lane = threadIdx.x;

    const int fm = lane & 15;
    const int fk = (lane >> 4) * 2;

    // B-fragment column = output pixel (fixed per lane)
    const int p  = pt * 16 + fm;
    const int oy = p / HW;
    const int ox = p - oy * HW;

    // A-fragment rows = selected channels (fixed per lane)
    const int c0 = idx[b * COUT + mt0 * 16 + fm];
    const int c1 = idx[b * COUT + mt1 * 16 + fm];
    const float* w0 = w + (size_t)c0 * KCONV;
    const float* w1 = w + (size_t)c1 * KCONV;
    const float* xb = x + (size_t)b * CIN * NPIX;

    v8f acc0 = {};
    v8f acc1 = {};

    #pragma unroll
    for (int tap = 0; tap < 9; ++tap) {
        const int kh = tap / 3;
        const int kw = tap - kh * 3;
        const int iy = oy + kh - 1;
        const int ix = ox + kw - 1;
        const bool valid = (iy >= 0) & (iy < HW) & (ix >= 0) & (ix < HW);
        const float* xtap = xb + iy * HW + ix;      // loads below are predicated

        #pragma unroll
        for (int s = 0; s < CIN / 4; ++s) {         // 16 WMMA K-steps over ci
            const int ci = 4 * s + fk;
            v2f bf, aA, aB;
            bf.x = valid ? xtap[(size_t) ci      * NPIX] : 0.0f;
            bf.y = valid ? xtap[(size_t)(ci + 1) * NPIX] : 0.0f;
            aA.x = w0[ ci      * 9 + tap];
            aA.y = w0[(ci + 1) * 9 + tap];
            aB.x = w1[ ci      * 9 + tap];
            aB.y = w1[(ci + 1) * 9 + tap];
            acc0 = __builtin_amdgcn_wmma_f32_16x16x4_f32(false, aA, false, bf,
                                                         (short)0, acc0, false, false);
            acc1 = __builtin_amdgcn_wmma_f32_16x16x4_f32(false, aB, false, bf,
                                                         (short)0, acc1, false, false);
        }
    }

    // bias + store: D layout M = v + 8*(lane>=16), N = lane&15
    const int mh = (lane >> 4) * 8;
    const int n  = lane & 15;
    #pragma unroll
    for (int v = 0; v < 8; ++v) {
        const int m   = v + mh;
        const int cm0 = idx[b * COUT + mt0 * 16 + m];
        const int cm1 = idx[b * COUT + mt1 * 16 + m];
        out[((size_t)(b * COUT + mt0 * 16 + m)) * NPIX + pt * 16 + n]
            = acc0[v] + bias[cm0];
        out[((size_t)(b * COUT + mt1 * 16 + m)) * NPIX + pt * 16 + n]
            = acc1[v] + bias[cm1];
    }
}

// ---------------------------------------------------------------------------
extern "C" void kernel_launch(void* const* d_in, const int* in_sizes, int n_in,
                              void* d_out, int out_size, void* d_ws, size_t ws_size,
                              hipStream_t stream) {
    const float* x        = (const float*)d_in[0];
    const float* conv_w   = (const float*)d_in[1];   // [4,128,64,3,3] == [512,64,3,3]
    const float* conv_b   = (const float*)d_in[2];   // [4,128] == [512]
    const float* router_w = (const float*)d_in[3];   // [512, 200704]
    const float* router_b = (const float*)d_in[4];   // [512]
    float* out = (float*)d_out;

    char*  ws      = (char*)d_ws;
    float* partial = (float*)ws;                                     // 16*64*512 f32 = 2 MB
    float* scores  = (float*)(ws + (size_t)KSPLIT * BATCH * CH * 4); // 64*512 f32
    int*   idx     = (int*)  (ws + (size_t)KSPLIT * BATCH * CH * 4
                                 + (size_t)BATCH * CH * 4);          // 64*128 i32

    dim3 g1(CH / 64, KSPLIT);
    router_partial<<<g1, 256, 0, stream>>>(x, router_w, partial);

    router_reduce<<<(BATCH * CH + 255) / 256, 256, 0, stream>>>(partial, router_b, scores);

    topk_kernel<<<BATCH, CH, 0, stream>>>(scores, idx);

    dim3 g3(NPIX / 16, COUT / 32, BATCH);
    conv_kernel<<<g3, 32, 0, stream>>>(x, conv_w, conv_b, idx, out);
}